// RollerPooling_77266461655553
// MI455X (gfx1250) — compile-verified
//
#include <hip/hip_runtime.h>
#include <hip/hip_bf16.h>
#include <stdint.h>

// RollerPooling: out[e,r] = mean of hidden[e, r : r+w) with w = 65 - target_size[e].
// Computed as suffix sums S = hidden x T (T lower-triangular ones, synthesized in
// registers) via V_WMMA_F32_16X16X4_F32, then windowed difference from LDS.

typedef __attribute__((ext_vector_type(2))) float v2f;
typedef __attribute__((ext_vector_type(8))) float v8f;

#define NCH 64
#define EDGES_PER_BLOCK 128
#define THREADS 256

__global__ __launch_bounds__(THREADS) void roller_pool_wmma_kernel(
    const float* __restrict__ hidden,
    const int*   __restrict__ target_size,
    float*       __restrict__ out)
{
    // 128 rows x 64 f32 = 32 KB. Phase 1: swizzled raw tile. Phase 3: reused for S.
    __shared__ float sh[EDGES_PER_BLOCK * NCH];

    const int t     = threadIdx.x;
    const int block = blockIdx.x;
    const uint32_t lds_base = (uint32_t)(uintptr_t)(&sh[0]);

    // ---------------- Phase 1: async global -> LDS (swizzled 16B groups) -------
    // chunk q (16B): row e = q>>4, group g = q&15; stored at group (g ^ (e&15)).
    {
        const float* gbase = hidden + (size_t)block * EDGES_PER_BLOCK * NCH;
        #pragma unroll
        for (int i = 0; i < 8; ++i) {
            int q = t + i * THREADS;                       // 0..2047
            int e = q >> 4;
            int g = q & 15;
            uint32_t lds_off = lds_base + (uint32_t)((e << 8) + (((g ^ (e & 15)) << 4)));
            int goff = q << 4;                             // global byte offset
            asm volatile("global_load_async_to_lds_b128 %0, %1, %2 offset:0"
                         :: "v"(lds_off), "v"(goff), "s"(gbase)
                         : "memory");
        }
    }
    asm volatile("s_wait_asynccnt 0" ::: "memory");
    __syncthreads();

    // ---------------- Phase 2: S = hidden_tile x T via WMMA f32 16x16x4 --------
    // Wave w owns edges [16w, 16w+16). A-matrix 16x4 layout (ISA 7.12.2):
    //   v0: lanes0-15 K=0 | lanes16-31 K=2 ; v1: K=1 | K=3  -> per lane a float2
    //   at channel c = 4k + 2*(lane>>4). B 4x16 mirrors: per-lane b = T[c..c+1][col],
    //   col = 16n + (lane&15);  T[c][col] = (c >= col).
    const int wave = t >> 5;
    const int lane = t & 31;
    const int r    = lane & 15;          // A row / B-C-D column sublane
    const int hi   = lane >> 4;
    const int wtile = wave * 16 * NCH;   // float index of this wave's 16 rows

    v8f acc[4];
    acc[0] = v8f{}; acc[1] = v8f{}; acc[2] = v8f{}; acc[3] = v8f{};

    #pragma unroll
    for (int k = 0; k < 16; ++k) {
        const int c = 4 * k + 2 * hi;
        // swizzled A read: row r, group (k ^ r), floats (2*hi, 2*hi+1) -> 8B aligned
        const int foff = wtile + r * NCH + ((k ^ r) << 2) + 2 * hi;
        v2f a = *(const v2f*)&sh[foff];

        #pragma unroll
        for (int n = 0; n < 4; ++n) {
            const int col = 16 * n + r;
            v2f b;
            b.x = (c     >= col) ? 1.0f : 0.0f;
            b.y = (c + 1 >= col) ? 1.0f : 0.0f;
            acc[n] = __builtin_amdgcn_wmma_f32_16x16x4_f32(
                false, a, false, b, (short)0, acc[n], false, false);
        }
    }

    // ---------------- Phase 3: spill S row-major into (dead) raw region --------
    // C/D layout: VGPR p -> M=p (lanes 0-15), M=p+8 (lanes 16-31); N = 16n + r.
    #pragma unroll
    for (int n = 0; n < 4; ++n) {
        #pragma unroll
        for (int p = 0; p < 8; ++p) {
            const int row = p + 8 * hi;
            sh[wtile + row * NCH + 16 * n + r] = acc[n][p];
        }
    }
    __syncthreads();

    // ---------------- Phase 4: windowed difference + coalesced store -----------
    // Per (wave, iter): single edge, 32 consecutive r -> both LDS reads conflict-free.
    float* outbase = out + (size_t)block * EDGES_PER_BLOCK * NCH;
    const int* tsbase = target_size + (size_t)block * EDGES_PER_BLOCK;
    #pragma unroll 4
    for (int i = 0; i < 32; ++i) {
        const int g  = t + i * THREADS;   // 0..8191
        const int e  = g >> 6;
        const int rr = g & 63;
        const int ts = tsbase[e];         // wave-uniform
        const int w  = NCH - ts + 1;      // window in [1, 64]
        const float s1 = sh[e * NCH + rr];
        const int   j  = rr + w;
        const float s2 = (j < NCH) ? sh[e * NCH + j] : 0.0f;
        outbase[g] = (s1 - s2) / (float)w;
    }
}

extern "C" void kernel_launch(void* const* d_in, const int* in_sizes, int n_in,
                              void* d_out, int out_size, void* d_ws, size_t ws_size,
                              hipStream_t stream) {
    const float* hidden      = (const float*)d_in[0];
    const int*   target_size = (const int*)d_in[1];
    // d_in[2] (pool_matrix) is not needed: T is synthesized analytically in registers.
    float* out = (float*)d_out;

    const int n_edges = in_sizes[0] / NCH;            // 65536
    const int blocks  = n_edges / EDGES_PER_BLOCK;    // 512
    roller_pool_wmma_kernel<<<blocks, THREADS, 0, stream>>>(hidden, target_size, out);
}